// HighOrderFactorizationMachineModel_8091718386298
// MI455X (gfx1250) — compile-verified
//
#include <hip/hip_runtime.h>

// High-order FM (order 3, closed forms):
//   y = bias + sum_f lin[idx_f]
//     + 0.5 * sum_{d<16}  (s1_d^2 - s2_d)                        // e2, cols 0..15
//     + (1/6) * sum_{d>=16}(s1_d^3 - 3 s1_d s2_d + 2 s3_d)       // e3, cols 16..31
// where s_k,d = sum_f emb[idx_f][d]^k.
//
// Gather path: CDNA5 async global->LDS loads (ASYNCcnt), one wave per row.

#define NUM_FIELDS   20
#define FIELD_DIM    50000
#define EMB_COLS     32          // EMBED_DIM * (ORDER-1)
#define WAVES_PER_BLOCK 8

__global__ __launch_bounds__(256) void hofm_kernel(
    const int*   __restrict__ x,        // (B, 20) int32
    const float* __restrict__ emb,      // (1e6, 32) f32
    const float* __restrict__ lin,      // (1e6, 1) f32
    const float* __restrict__ bias,     // (1,) f32
    float*       __restrict__ out,      // (B,) f32
    int B)
{
    __shared__ float smem[WAVES_PER_BLOCK][NUM_FIELDS][EMB_COLS];

    const int lane = threadIdx.x & 31;           // wave32
    const int w    = threadIdx.x >> 5;
    const int row  = blockIdx.x * WAVES_PER_BLOCK + w;
    if (row >= B) return;

    // Lanes 0..19: load this row's index for field `lane`, plus linear weight.
    int   idx    = 0;
    float linval = 0.0f;
    if (lane < NUM_FIELDS) {
        idx    = x[row * NUM_FIELDS + lane] + lane * FIELD_DIM;
        linval = lin[idx];
    }

    // Async gather of the 20x32 f32 tile into LDS:
    // 5 issues of global_load_async_to_lds_b128; per issue, 8 lanes x 16B
    // cover one field's 128B embedding row, 4 fields per issue.
    const int sub = lane & 7;                    // 16B chunk within a field row
    #pragma unroll
    for (int i = 0; i < 5; ++i) {
        const int field = i * 4 + (lane >> 3);   // 0..19
        const int fidx  = __shfl(idx, field, 32);
        unsigned long long gaddr =
            (unsigned long long)(const void*)(emb + (size_t)fidx * EMB_COLS + sub * 4);
        // Low 32 bits of the flat shared-aperture address == LDS byte offset.
        unsigned lds_off =
            (unsigned)(reinterpret_cast<uintptr_t>(&smem[w][field][sub * 4]));
        asm volatile("global_load_async_to_lds_b128 %0, %1, off"
                     :: "v"(lds_off), "v"(gaddr)
                     : "memory");
    }
    // Same-wave producer/consumer: split-counter wait, no barrier needed.
    asm volatile("s_wait_asynccnt 0" ::: "memory");

    // Lane c reduces embedding column c over the 20 fields.
    float s1 = 0.0f, s2 = 0.0f, s3 = 0.0f;
    #pragma unroll
    for (int f = 0; f < NUM_FIELDS; ++f) {
        const float v  = smem[w][f][lane];
        const float v2 = v * v;
        s1 += v;
        s2 += v2;
        s3 += v2 * v;
    }

    float contrib;
    if (lane < 16) {
        contrib = 0.5f * (s1 * s1 - s2);                              // order-2 (e2)
    } else {
        contrib = (s1 * s1 * s1 - 3.0f * s1 * s2 + 2.0f * s3)
                  * (1.0f / 6.0f);                                    // order-3 (e3)
    }
    contrib += linval;   // lanes 20..31 carry 0

    // Wave32 tree reduction.
    #pragma unroll
    for (int off = 16; off > 0; off >>= 1)
        contrib += __shfl_xor(contrib, off, 32);

    if (lane == 0)
        out[row] = contrib + bias[0];
}

extern "C" void kernel_launch(void* const* d_in, const int* in_sizes, int n_in,
                              void* d_out, int out_size, void* d_ws, size_t ws_size,
                              hipStream_t stream) {
    const int*   x    = (const int*)  d_in[0];   // (B, 20)
    const float* emb  = (const float*)d_in[1];   // (1e6, 32)
    const float* lin  = (const float*)d_in[2];   // (1e6, 1)
    const float* bias = (const float*)d_in[3];   // (1,)
    float*       out  = (float*)d_out;

    const int B = in_sizes[0] / NUM_FIELDS;
    const int blocks = (B + WAVES_PER_BLOCK - 1) / WAVES_PER_BLOCK;
    hofm_kernel<<<blocks, 256, 0, stream>>>(x, emb, lin, bias, out, B);
}